// MamMIL2GIA_29162827939922
// MI455X (gfx1250) — compile-verified
//
#include <hip/hip_runtime.h>

#define NN     4096
#define INDIM  1024
#define DM     192
#define DI     384
#define DSTATE 16
#define DTR    12
#define DCONV  4
#define EDG    32768
#define DBLW   44   // DT_RANK + 2*D_STATE
#define DBLP   48   // padded to multiple of 16
#define NSEQ   8

typedef float v2f __attribute__((ext_vector_type(2)));
typedef float v8f __attribute__((ext_vector_type(8)));

__device__ __forceinline__ float siluf(float x) { return x / (1.f + __expf(-x)); }
__device__ __forceinline__ float softplusf(float x) { return (x > 20.f) ? x : log1pf(__expf(x)); }

// order-preserving float<->uint key for atomicMax-based segment max
__device__ __forceinline__ unsigned fkey(float x) {
    unsigned b = __float_as_uint(x);
    return (b & 0x80000000u) ? ~b : (b | 0x80000000u);
}
__device__ __forceinline__ float fdec(unsigned k) {
    unsigned b = (k & 0x80000000u) ? (k & 0x7FFFFFFFu) : ~k;
    return __uint_as_float(b);
}

// ---------------------------------------------------------------------------
// fp32 WMMA GEMM:  C[M,N] = act(A[M,K] @ W[N,K]^T + bias)
// One wave32 computes a 32x16 macro-tile: two 16x16 V_WMMA_F32_16X16X4_F32
// accumulators sharing one B fragment per K-step.  Inner loop is branch-free
// (OOB W rows are pointer-clamped; their columns are never stored), so EXEC
// stays all-1 around the WMMAs and loads lower to plain global_load_b64.
// Requires: M % 32 == 0, N % 16 == 0, K % 4 == 0, lda/ldw even.
// ---------------------------------------------------------------------------
__global__ void __launch_bounds__(256) gemm_wmma(
    const float* __restrict__ A, int lda,
    const float* __restrict__ W, int ldw,
    const float* __restrict__ bias,
    float* __restrict__ C, int ldc,
    int M, int N, int Nreal, int K, int act)
{
    const int lane = threadIdx.x & 31;
    const int wave = blockIdx.x * (blockDim.x >> 5) + (threadIdx.x >> 5);
    const int tilesN = N >> 4;
    const int tm = (wave / tilesN) << 5;          // 32-row macro tile
    const int tn = (wave % tilesN) << 4;
    if (tm >= M) return;                          // wave-uniform exit
    const int hlf = lane >> 4;                    // 0 or 1
    const int l16 = lane & 15;
    const int rowW = tn + l16;
    const int rowWc = (rowW < Nreal) ? rowW : (Nreal - 1);   // clamp, cols unused
    const float* Ap0 = A + (size_t)(tm + l16) * lda;
    const float* Ap1 = A + (size_t)(tm + 16 + l16) * lda;
    const float* Wp  = W + (size_t)rowWc * ldw;

    v8f acc0 = {0.f, 0.f, 0.f, 0.f, 0.f, 0.f, 0.f, 0.f};
    v8f acc1 = {0.f, 0.f, 0.f, 0.f, 0.f, 0.f, 0.f, 0.f};
#pragma unroll 4
    for (int k0 = 0; k0 < K; k0 += 4) {
        const int ka = k0 + (hlf << 1);
        const v2f a0 = *(const v2f*)(Ap0 + ka);
        const v2f a1 = *(const v2f*)(Ap1 + ka);
        const v2f b  = *(const v2f*)(Wp + ka);
        acc0 = __builtin_amdgcn_wmma_f32_16x16x4_f32(false, a0, false, b,
                                                     (short)0, acc0, false, false);
        acc1 = __builtin_amdgcn_wmma_f32_16x16x4_f32(false, a1, false, b,
                                                     (short)0, acc1, false, false);
    }

    const int n = tn + l16;
    if (n < Nreal) {
        const float bv = bias ? bias[n] : 0.f;
#pragma unroll
        for (int j = 0; j < 8; ++j) {
            const int m = tm + j + (hlf << 3);
            float v0 = acc0[j] + bv;
            float v1 = acc1[j] + bv;
            if (act == 1)      { v0 = fmaxf(v0, 0.f);  v1 = fmaxf(v1, 0.f); }
            else if (act == 2) { v0 = tanhf(v0);       v1 = tanhf(v1); }
            else if (act == 3) { v0 = softplusf(v0);   v1 = softplusf(v1); }
            C[(size_t)m * ldc + n]        = v0;
            C[(size_t)(m + 16) * ldc + n] = v1;
        }
    }
}

// ---------------------------------------------------------------------------
// LayerNorm: one wave32 per row (D in {192,384}), optional fused ReLU.
// ---------------------------------------------------------------------------
__global__ void ln_rows(const float* __restrict__ src, float* __restrict__ dst,
                        const float* __restrict__ g, const float* __restrict__ b,
                        int D, int rows, int doRelu)
{
    const int warp = blockIdx.x * (blockDim.x >> 5) + (threadIdx.x >> 5);
    const int lane = threadIdx.x & 31;
    if (warp >= rows) return;
    const float* x = src + (size_t)warp * D;
    float s = 0.f;
    for (int i = lane; i < D; i += 32) s += x[i];
    for (int o = 16; o; o >>= 1) s += __shfl_xor(s, o, 32);
    const float mean = s / D;
    float v = 0.f;
    for (int i = lane; i < D; i += 32) { float t = x[i] - mean; v += t * t; }
    for (int o = 16; o; o >>= 1) v += __shfl_xor(v, o, 32);
    const float rstd = rsqrtf(v / D + 1e-5f);
    float* y = dst + (size_t)warp * D;
    for (int i = lane; i < D; i += 32) {
        float t = (x[i] - mean) * rstd * g[i] + b[i];
        if (doRelu) t = fmaxf(t, 0.f);
        y[i] = t;
    }
}

// ---------------------------------------------------------------------------
// Causal depthwise conv (win=4) + SiLU, gathering rows through the permutation
// for all 8 (perm, direction) sequences.  xz holds [xi | z] with ld = 768.
// ---------------------------------------------------------------------------
__global__ void conv_silu(const float* __restrict__ xz,
                          const int* __restrict__ p1, const int* __restrict__ p2,
                          const int* __restrict__ p3,
                          const float* __restrict__ cw, const float* __restrict__ cb,
                          float* __restrict__ xc)
{
    const int d = blockIdx.x * blockDim.x + threadIdx.x;
    if (d >= DI) return;
    const int t = blockIdx.y;
    const int q = blockIdx.z;
    const int pi = q >> 1, rev = q & 1;
    const int* pp = (pi == 1) ? p1 : (pi == 2) ? p2 : p3;
    float acc = cb[d];
#pragma unroll
    for (int k = 0; k < DCONV; ++k) {
        const int u = t - (DCONV - 1) + k;
        if (u < 0) continue;
        const int up = rev ? (NN - 1 - u) : u;
        const int r = (pi == 0) ? up : pp[up];
        acc += xz[(size_t)r * (2 * DI) + d] * cw[d * DCONV + k];
    }
    xc[((size_t)q * NN + t) * DI + d] = siluf(acc);
}

// ---------------------------------------------------------------------------
// Selective scan: one block per sequence, one thread per channel.
// dt_proj (K=12) fused in registers; per-step dbl row (dt_in|B|C) double-
// buffered through LDS; output scattered via forward perm with atomicAdd.
// ---------------------------------------------------------------------------
__global__ void __launch_bounds__(DI) mamba_scan(
    const float* __restrict__ dbl, const float* __restrict__ xc,
    const float* __restrict__ A_log, const float* __restrict__ dtw,
    const float* __restrict__ dtb, const float* __restrict__ Dp,
    const int* __restrict__ p1, const int* __restrict__ p2,
    const int* __restrict__ p3,
    float* __restrict__ ysum)
{
    const int d = threadIdx.x;     // 0..383
    const int q = blockIdx.x;      // 0..7
    const int pi = q >> 1, rev = q & 1;
    const int* pp = (pi == 1) ? p1 : (pi == 2) ? p2 : p3;

    float As[DSTATE];
#pragma unroll
    for (int s = 0; s < DSTATE; ++s) As[s] = -__expf(A_log[d * DSTATE + s]);
    float w[DTR];
#pragma unroll
    for (int r = 0; r < DTR; ++r) w[r] = dtw[d * DTR + r];
    const float bt = dtb[d], Dd = Dp[d];
    float st[DSTATE];
#pragma unroll
    for (int s = 0; s < DSTATE; ++s) st[s] = 0.f;

    __shared__ float sh[2][DBLP];
    if (d < DBLP) sh[0][d] = dbl[((size_t)q * NN) * DBLP + d];
    __syncthreads();

    for (int t = 0; t < NN; ++t) {
        const float* row = sh[t & 1];
        if (d < DBLP && (t + 1) < NN)
            sh[(t + 1) & 1][d] = dbl[((size_t)q * NN + t + 1) * DBLP + d];

        float di = bt;
#pragma unroll
        for (int r = 0; r < DTR; ++r) di += row[r] * w[r];
        const float dt = softplusf(di);
        const float x = xc[((size_t)q * NN + t) * DI + d];
        const float dtx = dt * x;
        float y = 0.f;
#pragma unroll
        for (int s = 0; s < DSTATE; ++s) {
            st[s] = st[s] * __expf(dt * As[s]) + dtx * row[DTR + s];
            y += st[s] * row[DTR + DSTATE + s];
        }
        y += x * Dd;
        const int up = rev ? (NN - 1 - t) : t;
        const int n = (pi == 0) ? up : pp[up];
        atomicAdd(&ysum[(size_t)n * DI + d], y);
        __syncthreads();
    }
}

// ---------------------------------------------------------------------------
// Small elementwise / reduction kernels
// ---------------------------------------------------------------------------
__global__ void fill_u32(unsigned* __restrict__ p, unsigned v, int n)
{
    const int i = blockIdx.x * blockDim.x + threadIdx.x;
    if (i < n) p[i] = v;
}

__global__ void ygate(float* __restrict__ ysum, const float* __restrict__ xz, int n)
{
    const int i = blockIdx.x * blockDim.x + threadIdx.x;
    if (i >= n) return;
    const int nr = i / DI, d = i - nr * DI;
    const float z = xz[(size_t)nr * (2 * DI) + DI + d];
    ysum[i] = ysum[i] * 0.125f * siluf(z);
}

__global__ void __launch_bounds__(DM) edge_max(
    const int* __restrict__ ei, const float* __restrict__ h1,
    const float* __restrict__ tptr, unsigned* __restrict__ mxk)
{
    const int f = threadIdx.x;
    const int e = blockIdx.x;
    const int s = ei[e], dn = ei[EDG + e];
    const float msg = fmaxf(h1[(size_t)s * DM + f], 0.f) + 1e-7f;
    const float logit = tptr[0] * msg;
    atomicMax(&mxk[(size_t)dn * DM + f], fkey(logit));
}

__global__ void mx_fin(unsigned* __restrict__ mx, int n)
{
    const int i = blockIdx.x * blockDim.x + threadIdx.x;
    if (i >= n) return;
    float v = fdec(mx[i]);
    const unsigned eb = (__float_as_uint(v) >> 23) & 0xFFu;
    if (eb == 0xFFu) v = 0.f;          // non-finite -> 0
    ((float*)mx)[i] = v;
}

__global__ void __launch_bounds__(DM) edge_sum(
    const int* __restrict__ ei, const float* __restrict__ h1,
    const float* __restrict__ tptr, const float* __restrict__ mx,
    float* __restrict__ den, float* __restrict__ num)
{
    const int f = threadIdx.x;
    const int e = blockIdx.x;
    const int s = ei[e], dn = ei[EDG + e];
    const float msg = fmaxf(h1[(size_t)s * DM + f], 0.f) + 1e-7f;
    const float logit = tptr[0] * msg;
    const float ex = __expf(logit - mx[(size_t)dn * DM + f]);
    atomicAdd(&den[(size_t)dn * DM + f], ex);
    atomicAdd(&num[(size_t)dn * DM + f], ex * msg);
}

__global__ void u_final(const float* __restrict__ num, const float* __restrict__ den,
                        const float* __restrict__ h1, float* __restrict__ u, int n)
{
    const int i = blockIdx.x * blockDim.x + threadIdx.x;
    if (i >= n) return;
    const float dd = den[i];
    u[i] = num[i] / (dd > 0.f ? dd : 1.f) + h1[i];
}

__global__ void add3(const float* __restrict__ a, const float* __restrict__ b,
                     const float* __restrict__ c, float* __restrict__ o, int n)
{
    const int i = blockIdx.x * blockDim.x + threadIdx.x;
    if (i < n) o[i] = a[i] + b[i] + c[i];
}

__global__ void attn_score(const float* __restrict__ a1, const float* __restrict__ w2,
                           const float* __restrict__ b2, float* __restrict__ out)
{
    const int warp = blockIdx.x * (blockDim.x >> 5) + (threadIdx.x >> 5);
    const int lane = threadIdx.x & 31;
    if (warp >= NN) return;
    const float* x = a1 + (size_t)warp * 128;
    float s = 0.f;
#pragma unroll
    for (int i = lane; i < 128; i += 32) s += x[i] * w2[i];
    for (int o = 16; o; o >>= 1) s += __shfl_xor(s, o, 32);
    if (lane == 0) out[warp] = s + b2[0];
}

__global__ void __launch_bounds__(1024) softmax_n(const float* __restrict__ a,
                                                  float* __restrict__ w)
{
    __shared__ float red[32];
    __shared__ float red2[32];
    const int tid = threadIdx.x;
    float mx = -1e30f;
    for (int i = tid; i < NN; i += 1024) mx = fmaxf(mx, a[i]);
    for (int o = 16; o; o >>= 1) mx = fmaxf(mx, __shfl_xor(mx, o, 32));
    if ((tid & 31) == 0) red[tid >> 5] = mx;
    __syncthreads();
    if (tid < 32) {
        float v = red[tid];
        for (int o = 16; o; o >>= 1) v = fmaxf(v, __shfl_xor(v, o, 32));
        red[tid] = v;
    }
    __syncthreads();
    mx = red[0];
    float s = 0.f;
    for (int i = tid; i < NN; i += 1024) s += __expf(a[i] - mx);
    for (int o = 16; o; o >>= 1) s += __shfl_xor(s, o, 32);
    if ((tid & 31) == 0) red2[tid >> 5] = s;
    __syncthreads();
    if (tid < 32) {
        float v = red2[tid];
        for (int o = 16; o; o >>= 1) v += __shfl_xor(v, o, 32);
        red2[tid] = v;
    }
    __syncthreads();
    const float inv = 1.f / red2[0];
    for (int i = tid; i < NN; i += 1024) w[i] = __expf(a[i] - mx) * inv;
}

__global__ void __launch_bounds__(256) pool_feat(const float* __restrict__ w,
                                                 const float* __restrict__ hn2,
                                                 float* __restrict__ pooled)
{
    const int f = blockIdx.x;
    const int tid = threadIdx.x;
    float s = 0.f;
    for (int i = tid; i < NN; i += 256) s += w[i] * hn2[(size_t)i * DM + f];
    for (int o = 16; o; o >>= 1) s += __shfl_xor(s, o, 32);
    __shared__ float red[8];
    if ((tid & 31) == 0) red[tid >> 5] = s;
    __syncthreads();
    if (tid == 0) {
        float t = 0.f;
        for (int j = 0; j < 8; ++j) t += red[j];
        pooled[f] = t;
    }
}

__global__ void head_k(const float* __restrict__ pooled, const float* __restrict__ hw,
                       const float* __restrict__ hb, float* __restrict__ out)
{
    const int c = threadIdx.x >> 5;      // blockDim = 64 -> c in {0,1}
    const int lane = threadIdx.x & 31;
    float s = 0.f;
    for (int i = lane; i < DM; i += 32) s += pooled[i] * hw[c * DM + i];
    for (int o = 16; o; o >>= 1) s += __shfl_xor(s, o, 32);
    if (lane == 0) out[c] = s + hb[c];
}

// ---------------------------------------------------------------------------
extern "C" void kernel_launch(void* const* d_in, const int* in_sizes, int n_in,
                              void* d_out, int out_size, void* d_ws, size_t ws_size,
                              hipStream_t stream)
{
    (void)in_sizes; (void)n_in; (void)out_size; (void)ws_size;
    const float* x      = (const float*)d_in[0];
    const int*   ei     = (const int*)  d_in[1];
    const int*   p1     = (const int*)  d_in[2];
    const int*   p2     = (const int*)  d_in[3];
    const int*   p3     = (const int*)  d_in[4];
    const float* ft_w   = (const float*)d_in[5];
    const float* ft_b   = (const float*)d_in[6];
    const float* ln1_g  = (const float*)d_in[7];
    const float* ln1_b  = (const float*)d_in[8];
    const float* inp_w  = (const float*)d_in[9];
    const float* conv_w = (const float*)d_in[10];
    const float* conv_b = (const float*)d_in[11];
    const float* xpw    = (const float*)d_in[12];
    const float* dtw    = (const float*)d_in[13];
    const float* dtb    = (const float*)d_in[14];
    const float* A_log  = (const float*)d_in[15];
    const float* Dp     = (const float*)d_in[16];
    const float* outp_w = (const float*)d_in[17];
    const float* gnn_t  = (const float*)d_in[18];
    const float* m1w    = (const float*)d_in[19];
    const float* m1b    = (const float*)d_in[20];
    const float* mlng   = (const float*)d_in[21];
    const float* mlnb   = (const float*)d_in[22];
    const float* m2w    = (const float*)d_in[23];
    const float* m2b    = (const float*)d_in[24];
    const float* gng    = (const float*)d_in[25];
    const float* gnb    = (const float*)d_in[26];
    const float* ng     = (const float*)d_in[27];
    const float* nb     = (const float*)d_in[28];
    const float* a1w    = (const float*)d_in[29];
    const float* a1b    = (const float*)d_in[30];
    const float* a2w    = (const float*)d_in[31];
    const float* a2b    = (const float*)d_in[32];
    const float* hw     = (const float*)d_in[33];
    const float* hb     = (const float*)d_in[34];
    float* out = (float*)d_out;

    float* ws = (float*)d_ws;
    size_t off = 0;
    auto alloc = [&](size_t n) { float* p = ws + off; off += n; return p; };
    float* h0   = alloc((size_t)NN * DM);
    float* hn   = alloc((size_t)NN * DM);
    float* xzb  = alloc((size_t)NN * 2 * DI);
    float* xc   = alloc((size_t)NSEQ * NN * DI);
    float* dbl  = alloc((size_t)NSEQ * NN * DBLP);
    float* ysum = alloc((size_t)NN * DI);
    float* h1   = alloc((size_t)NN * DM);
    float* mx   = alloc((size_t)NN * DM);   // uint keys, then floats in place
    float* den  = alloc((size_t)NN * DM);
    float* num  = alloc((size_t)NN * DM);
    float* u    = alloc((size_t)NN * DM);
    float* m1   = alloc((size_t)NN * 2 * DM);
    float* gout = alloc((size_t)NN * DM);
    float* h3   = alloc((size_t)NN * DM);
    float* hn2  = alloc((size_t)NN * DM);
    float* a1   = alloc((size_t)NN * 128);
    float* asc  = alloc(NN);
    float* wsm  = alloc(NN);
    float* pooled = alloc(DM);

    auto gemm = [&](const float* A, int lda, const float* W, int ldw, const float* bias,
                    float* C, int ldc, int M, int Nc, int Nreal, int K, int act) {
        const int tiles = (M >> 5) * (Nc >> 4);        // 32x16 macro-tiles
        const int blocks = (tiles + 7) / 8;            // 8 waves / 256-thread block
        gemm_wmma<<<blocks, 256, 0, stream>>>(A, lda, W, ldw, bias, C, ldc,
                                              M, Nc, Nreal, K, act);
    };
    const int NM = NN * DM, NI = NN * DI;

    // 1. h0 = relu(x @ ft_w^T + ft_b)
    gemm(x, INDIM, ft_w, INDIM, ft_b, h0, DM, NN, DM, DM, INDIM, 1);
    // 2. hn = LN(h0)
    ln_rows<<<(NN * 32 + 255) / 256, 256, 0, stream>>>(h0, hn, ln1_g, ln1_b, DM, NN, 0);
    // 3. xz = hn @ in_proj_w^T
    gemm(hn, DM, inp_w, DM, nullptr, xzb, 2 * DI, NN, 2 * DI, 2 * DI, DM, 0);
    // 4. xc[q,t,d] = silu(conv(permuted xi))
    conv_silu<<<dim3(3, NN, NSEQ), 128, 0, stream>>>(xzb, p1, p2, p3, conv_w, conv_b, xc);
    // 5. dbl = xc @ x_proj_w^T   (N=44 padded to 48; OOB W rows pointer-clamped)
    gemm(xc, DI, xpw, DI, nullptr, dbl, DBLP, NSEQ * NN, DBLP, DBLW, DI, 0);
    // 6. selective scan with fused dt_proj + scatter-accumulate
    fill_u32<<<(NI + 255) / 256, 256, 0, stream>>>((unsigned*)ysum, 0u, NI);
    mamba_scan<<<NSEQ, DI, 0, stream>>>(dbl, xc, A_log, dtw, dtb, Dp, p1, p2, p3, ysum);
    // 7. y = (ysum/8) * silu(z)
    ygate<<<(NI + 255) / 256, 256, 0, stream>>>(ysum, xzb, NI);
    // 8. h1 = y @ out_proj_w^T
    gemm(ysum, DI, outp_w, DI, nullptr, h1, DM, NN, DM, DM, DI, 0);
    // 9. GNN segment softmax aggregation
    fill_u32<<<(NM + 255) / 256, 256, 0, stream>>>((unsigned*)mx, 0x007FFFFFu, NM); // key(-inf)
    fill_u32<<<(NM + 255) / 256, 256, 0, stream>>>((unsigned*)den, 0u, NM);
    fill_u32<<<(NM + 255) / 256, 256, 0, stream>>>((unsigned*)num, 0u, NM);
    edge_max<<<EDG, DM, 0, stream>>>(ei, h1, gnn_t, (unsigned*)mx);
    mx_fin<<<(NM + 255) / 256, 256, 0, stream>>>((unsigned*)mx, NM);
    edge_sum<<<EDG, DM, 0, stream>>>(ei, h1, gnn_t, mx, den, num);
    u_final<<<(NM + 255) / 256, 256, 0, stream>>>(num, den, h1, u, NM);
    // 10. MLP with LN+relu
    gemm(u, DM, m1w, DM, m1b, m1, 2 * DM, NN, 2 * DM, 2 * DM, DM, 0);
    ln_rows<<<(NN * 32 + 255) / 256, 256, 0, stream>>>(m1, m1, mlng, mlnb, 2 * DM, NN, 1);
    gemm(m1, 2 * DM, m2w, 2 * DM, m2b, gout, DM, NN, DM, DM, 2 * DM, 0);
    ln_rows<<<(NN * 32 + 255) / 256, 256, 0, stream>>>(gout, gout, gng, gnb, DM, NN, 1);
    // 11. residuals + final LN
    add3<<<(NM + 255) / 256, 256, 0, stream>>>(h1, gout, h0, h3, NM);
    ln_rows<<<(NN * 32 + 255) / 256, 256, 0, stream>>>(h3, hn2, ng, nb, DM, NN, 0);
    // 12. attention pooling + head
    gemm(hn2, DM, a1w, DM, a1b, a1, 128, NN, 128, 128, DM, 2);
    attn_score<<<(NN * 32 + 255) / 256, 256, 0, stream>>>(a1, a2w, a2b, asc);
    softmax_n<<<1, 1024, 0, stream>>>(asc, wsm);
    pool_feat<<<DM, 256, 0, stream>>>(wsm, hn2, pooled);
    head_k<<<1, 64, 0, stream>>>(pooled, hw, hb, out);
}